// MiniRocket_38809324487413
// MI455X (gfx1250) — compile-verified
//
#include <hip/hip_runtime.h>

typedef float v2f __attribute__((ext_vector_type(2)));
typedef float v8f __attribute__((ext_vector_type(8)));

#define NKERN 84
#define MPAD  96
#define KDIM  72          // 8 channels * 9 taps
#define LEN   1024
#define NCH   8
#define NBATCH 64
#define OUTF  9996
#define XPITCH 1088       // 32 zero pad + 1024 + 32 zero pad
#define XPAD   32

// ---------------------------------------------------------------------------
// Kernel 1: build fused weight matrices Wbig[4][96][72]
// ---------------------------------------------------------------------------
__global__ void mr_build_wbig(const int* __restrict__ ci, float* __restrict__ wbig) {
    int idx = blockIdx.x * blockDim.x + threadIdx.x;
    if (idx >= 4 * MPAD * KDIM) return;
    int i   = idx / (MPAD * KDIM);
    int rem = idx % (MPAD * KDIM);
    int j   = rem / KDIM;
    int r   = rem % KDIM;
    float val = 0.0f;
    if (j < NKERN) {
        int c = r / 9, k = r % 9;
        const int* row = ci + (i * NKERN + j) * 3;
        bool chon = (row[0] == c) | (row[1] == c) | (row[2] == c);
        if (chon) {
            int a0 = 0, a1 = 1, a2 = 2, cnt = 0;
            for (int a = 0; a < 7; ++a)
                for (int b = a + 1; b < 8; ++b)
                    for (int cc = b + 1; cc < 9; ++cc) {
                        if (cnt == j) { a0 = a; a1 = b; a2 = cc; }
                        ++cnt;
                    }
            bool tap = (k == a0) | (k == a1) | (k == a2);
            val = tap ? 2.0f : -1.0f;
        }
    }
    wbig[idx] = val;
}

// ---------------------------------------------------------------------------
// Kernel 2: sort biases ascending per (dilation, kernel), keep permutation.
// ---------------------------------------------------------------------------
__global__ void mr_sort_bias(const float* __restrict__ biases,
                             float* __restrict__ sb, int* __restrict__ pm) {
    int idx = blockIdx.x * blockDim.x + threadIdx.x;
    if (idx >= 4 * NKERN) return;
    int i = idx / NKERN;
    int nf = (i == 3) ? 29 : 30;
    int j = idx % NKERN;
    int boff = 2520 * i;

    float v[30]; int px[30];
    for (int f = 0; f < nf; ++f) { v[f] = biases[boff + j * nf + f]; px[f] = f; }
    for (int a = 1; a < nf; ++a) {               // insertion sort ascending
        float key = v[a]; int kp = px[a]; int b = a - 1;
        while (b >= 0 && v[b] > key) { v[b + 1] = v[b]; px[b + 1] = px[b]; --b; }
        v[b + 1] = key; px[b + 1] = kp;
    }
    float* so = sb + idx * 32;
    int*   po = pm + idx * 32;
    for (int s = 0; s < 32; ++s) {
        so[s] = (s < nf) ? v[s] : __builtin_inff();
        po[s] = (s < nf) ? px[s] : 0;
    }
}

// ---------------------------------------------------------------------------
// Kernel 3: fused GEMM (fp32 WMMA 16x16x4) + rank-histogram PPV.
//   grid = 256 (= 64 batches x 4 dilations), block = 192 (6 waves).
// ---------------------------------------------------------------------------
__launch_bounds__(192)
__global__ void mr_wmma_ppv(const float* __restrict__ x,
                            const float* __restrict__ wbig,
                            const float* __restrict__ sb,
                            const int* __restrict__ pm,
                            float* __restrict__ out) {
    __shared__ __align__(16) float sx[NCH * XPITCH];   // 34816 B
    __shared__ float    ssb[MPAD * 32];                // 12288 B sorted biases
    __shared__ unsigned shist[MPAD * 32];              // 12288 B rank histograms

    const int wg = blockIdx.x;
    const int n  = wg >> 2;          // batch
    const int i  = wg & 3;           // dilation index
    const int d  = 1 << i;
    const int p  = 4 * d;
    const int nf = (i == 3) ? 29 : 30;
    const int boff = 2520 * i;

    // ---- stage x[n] (zero aprons), sorted biases, zero histograms -----------
    for (int q = threadIdx.x; q < NCH * 2 * XPAD; q += blockDim.x) {
        int c = q / (2 * XPAD);
        int w = q % (2 * XPAD);
        int tt = (w < XPAD) ? w : (XPITCH - 2 * XPAD + w);
        sx[c * XPITCH + tt] = 0.0f;
    }
    const float4* xg4 = (const float4*)(x + n * (NCH * LEN));
    for (int q = threadIdx.x; q < NCH * (LEN / 4); q += blockDim.x) {
        int c = q >> 8;              // LEN/4 == 256
        int w = q & 255;
        *(float4*)&sx[c * XPITCH + XPAD + w * 4] = xg4[c * 256 + w];
    }
    for (int q = threadIdx.x; q < MPAD * 32; q += blockDim.x) {
        int j = q >> 5, s = q & 31;
        ssb[q]   = (j < NKERN) ? sb[(i * NKERN + j) * 32 + s] : __builtin_inff();
        shist[q] = 0u;
    }
    __syncthreads();

    const int wave = threadIdx.x >> 5;   // 0..5 == M-tile index
    const int lane = threadIdx.x & 31;
    const int hl   = lane >> 4;          // half-wave
    const int l15  = lane & 15;
    const int J0   = wave * 16;

    const float* wb = wbig + i * (MPAD * KDIM);

    // ---- resident A fragments: 18 K-chunks of 4.  A(16x4): m=l15, k=v+2*hl --
    v2f afrag[18];
#pragma unroll
    for (int kk = 0; kk < 18; ++kk) {
        const float* src = wb + (J0 + l15) * KDIM + kk * 4 + 2 * hl;
        v2f a = { src[0], src[1] };
        afrag[kk] = a;
    }

    // ---- nt-invariant per-lane LDS offsets for the 36 B elements ------------
    // B(4x16): n=l15, k=v+2*hl ; row r -> channel r/9, tap r%9
    int boffs[36];
#pragma unroll
    for (int kk = 0; kk < 18; ++kk) {
        int r0 = kk * 4 + 2 * hl;
        int r1 = r0 + 1;
        boffs[2 * kk]     = (r0 / 9) * XPITCH + XPAD + ((r0 % 9) - 4) * d + l15;
        boffs[2 * kk + 1] = (r1 / 9) * XPITCH + XPAD + ((r1 % 9) - 4) * d + l15;
    }

    // ---- preload binary-search levels 1-2 into registers (nt-invariant) -----
    float med1[8], med2lo[8], med2hi[8];
#pragma unroll
    for (int v = 0; v < 8; ++v) {
        const float* sr = ssb + (J0 + v + 8 * hl) * 32;
        med1[v]   = sr[15];
        med2lo[v] = sr[7];
        med2hi[v] = sr[23];
    }

    for (int nt = 0; nt < LEN / 16; ++nt) {
        const int tbase = nt * 16;
        const int t0n   = tbase + l15;

        // gather all 36 B elements so ds_loads pipeline ahead of the WMMAs
        v2f bfrag[18];
#pragma unroll
        for (int kk = 0; kk < 18; ++kk) {
            bfrag[kk].x = sx[boffs[2 * kk]     + tbase];
            bfrag[kk].y = sx[boffs[2 * kk + 1] + tbase];
        }

        v8f acc = {};
#pragma unroll
        for (int kk = 0; kk < 18; ++kk) {
            acc = __builtin_amdgcn_wmma_f32_16x16x4_f32(
                false, afrag[kk], false, bfrag[kk], (short)0, acc, false, false);
        }

        // ---- rank-histogram PPV on the 16x16 tile ---------------------------
        const bool interior = (t0n >= p) && (t0n < LEN - p);
#pragma unroll
        for (int v = 0; v < 8; ++v) {
            // j = J0 + v + 8*hl ; parity(i+j) == parity(i+v) (uniform per v)
            const bool evenpar = (((i + v) & 1) == 0);
            const int jsel = J0 + v + 8 * hl;
            const float cv = acc[v];
            const float* sr = ssb + jsel * 32;
            // branchless lower-bound: r = #biases < cv  (r <= nf <= 30)
            const bool g1 = (med1[v] < cv);
            int r = g1 ? 16 : 0;
            const float l2 = g1 ? med2hi[v] : med2lo[v];
            if (l2        < cv) r += 8;
            if (sr[r + 3] < cv) r += 4;
            if (sr[r + 1] < cv) r += 2;
            if (sr[r]     < cv) r += 1;
            if (evenpar || interior)
                atomicAdd(&shist[jsel * 32 + r], 1u);   // wave-private j rows
        }
    }

    // ---- suffix-sum histograms -> counts, scatter through permutation -------
    if (lane < nf) {
        const int f = lane;                       // sorted position
        float* o = out + n * OUTF + boff;
#pragma unroll
        for (int v = 0; v < 8; ++v) {
            const bool evenpar = (((i + v) & 1) == 0);
            const float inv = evenpar ? (1.0f / (float)LEN)
                                      : (1.0f / (float)(LEN - 2 * p));
            const int jlo = J0 + v;
            const int jhi = J0 + 8 + v;
            if (jlo < NKERN) {
                unsigned cnt = 0;
                for (int r = f + 1; r <= nf; ++r) cnt += shist[jlo * 32 + r];
                int forig = pm[(i * NKERN + jlo) * 32 + f];
                o[jlo * nf + forig] = (float)cnt * inv;
            }
            if (jhi < NKERN) {
                unsigned cnt = 0;
                for (int r = f + 1; r <= nf; ++r) cnt += shist[jhi * 32 + r];
                int forig = pm[(i * NKERN + jhi) * 32 + f];
                o[jhi * nf + forig] = (float)cnt * inv;
            }
        }
    }
}

// ---------------------------------------------------------------------------
extern "C" void kernel_launch(void* const* d_in, const int* in_sizes, int n_in,
                              void* d_out, int out_size, void* d_ws, size_t ws_size,
                              hipStream_t stream) {
    const float* x      = (const float*)d_in[0];   // (64, 8, 1024) f32
    const float* biases = (const float*)d_in[1];   // (9996,)       f32
    const int*   ci     = (const int*)d_in[2];     // (336, 3)      i32
    float*       out    = (float*)d_out;           // (64, 9996)    f32

    float* wbig = (float*)d_ws;                    // 4*96*72      floats
    float* sb   = wbig + 4 * MPAD * KDIM;          // 4*84*32      floats
    int*   pm   = (int*)(sb + 4 * NKERN * 32);     // 4*84*32      ints

    const int tot = 4 * MPAD * KDIM;
    hipLaunchKernelGGL(mr_build_wbig, dim3((tot + 255) / 256), dim3(256), 0, stream,
                       ci, wbig);
    hipLaunchKernelGGL(mr_sort_bias, dim3(2), dim3(256), 0, stream,
                       biases, sb, pm);
    hipLaunchKernelGGL(mr_wmma_ppv, dim3(NBATCH * 4), dim3(192), 0, stream,
                       x, wbig, sb, pm, out);
}